// Reg_version_wave_40570261078380
// MI455X (gfx1250) — compile-verified
//
#include <hip/hip_runtime.h>

#define N_DIM 1024
#define B_DIM 64

typedef __attribute__((ext_vector_type(2))) float v2f;
typedef __attribute__((ext_vector_type(8))) float v8f;

// ---------------------------------------------------------------------------
// Kernel 1: partial diagonal sums.
// waves[b,d] needs sum_i attns[b, i, i+d].  Threads are indexed by d so that
// lanes read consecutive addresses (i*N + i + d, d contiguous) -> coalesced.
// The i-range is split into `isplit` chunks (blockIdx.y) for parallelism;
// partials go to workspace (no atomics -> bitwise deterministic).
// ---------------------------------------------------------------------------
__global__ void diag_partial_kernel(const float* __restrict__ attns,
                                    float* __restrict__ part,
                                    int isplit, int ilen) {
  const int d  = blockIdx.x * blockDim.x + threadIdx.x;   // diagonal offset
  const int iz = blockIdx.y;                              // i-chunk
  const int b  = blockIdx.z;                              // batch
  const int i0 = iz * ilen;
  int iend = i0 + ilen;
  const int lim = N_DIM - d;                              // need i + d < N
  if (iend > lim) iend = lim;
  const float* base = attns + (size_t)b * N_DIM * N_DIM;
  float s = 0.0f;
  for (int i = i0; i < iend; ++i)
    s += base[(size_t)i * N_DIM + i + d];
  part[((size_t)b * isplit + iz) * N_DIM + d] = s;
}

// ---------------------------------------------------------------------------
// Kernel 2: combine i-chunk partials and divide by diagonal length (N-d).
// ---------------------------------------------------------------------------
__global__ void waves_finalize_kernel(const float* __restrict__ part,
                                      float* __restrict__ waves, int isplit) {
  const int d = blockIdx.x * blockDim.x + threadIdx.x;
  const int b = blockIdx.y;
  float s = 0.0f;
  for (int iz = 0; iz < isplit; ++iz)
    s += part[((size_t)b * isplit + iz) * N_DIM + d];
  waves[b * N_DIM + d] = s / (float)(N_DIM - d);
}

// ---------------------------------------------------------------------------
// Kernel 3: DFT magnitude via V_WMMA_F32_16X16X4_F32.
//   spec[b,k] = | sum_d waves[b,d] * exp(-2*pi*i*k*d/N) |,  k = 1..512
// One wave per 16x16 output tile: tileM in [0,4) over batches, tileN in
// [0,32) over frequencies (32*16 = 512 bins exactly).  Twiddles computed
// exactly via (d*k mod N) so v_sin/v_cos see args in [0, 2*pi).
// A-matrix layout (16x4 f32, ISA 7.12.2): lane L holds M = L%16;
//   VGPR0 = K = 2*(L/16), VGPR1 = K = 2*(L/16)+1.  B mirrors the K striping.
// Per-tile max/sum of magnitudes reduced with shfl_xor over each 16-lane
// half (one half per 8 batch rows), written per tile -> deterministic.
// ---------------------------------------------------------------------------
__global__ void dft_wmma_kernel(const float* __restrict__ waves,
                                float* __restrict__ partmax,
                                float* __restrict__ partsum) {
  const int lane  = threadIdx.x & 31;
  const int half  = lane >> 4;
  const int l16   = lane & 15;
  const int tileM = blockIdx.x & 3;    // 4 tiles over 64 batches
  const int tileN = blockIdx.x >> 2;   // 32 tiles over 512 freq bins
  const int m = tileM * 16 + l16;      // batch row this lane loads for A
  const int k = 1 + tileN * 16 + l16;  // frequency this lane supplies for B
  const float scale = 6.2831853071795864769f / (float)N_DIM;

  v8f cre = {};  // real accumulator (16x16 f32 tile)
  v8f cim = {};  // imag accumulator (sign irrelevant for magnitude)

  for (int k0 = 0; k0 < N_DIM; k0 += 4) {
    const int dA = k0 + 2 * half;
    v2f a;
    a.x = waves[m * N_DIM + dA];
    a.y = waves[m * N_DIM + dA + 1];
    const int p0 = (dA * k) & (N_DIM - 1);
    const int p1 = ((dA + 1) * k) & (N_DIM - 1);
    const float t0 = (float)p0 * scale;
    const float t1 = (float)p1 * scale;
    v2f bc, bs;
    bc.x = __cosf(t0); bc.y = __cosf(t1);
    bs.x = __sinf(t0); bs.y = __sinf(t1);
    // 8 args: (neg_a, A, neg_b, B, c_mod, C, reuse_a, reuse_b)
    cre = __builtin_amdgcn_wmma_f32_16x16x4_f32(false, a, false, bc,
                                                (short)0, cre, false, false);
    cim = __builtin_amdgcn_wmma_f32_16x16x4_f32(false, a, false, bs,
                                                (short)0, cim, false, false);
  }

  // D layout: VGPR j, lanes 0-15 -> M=j, N=l16; lanes 16-31 -> M=j+8, N=l16.
  // For fixed (j, half) all 16 lanes of the half share a batch row and span
  // 16 frequencies -> reduce across the half with xor-shuffles.
  for (int j = 0; j < 8; ++j) {
    const float re = cre[j], im = cim[j];
    float v = sqrtf(re * re + im * im);
    float vmax = v, vsum = v;
#pragma unroll
    for (int off = 1; off < 16; off <<= 1) {
      vmax = fmaxf(vmax, __shfl_xor(vmax, off, 32));
      vsum += __shfl_xor(vsum, off, 32);
    }
    if (l16 == 0) {
      const int bM = tileM * 16 + j + 8 * half;
      partmax[bM * 32 + tileN] = vmax;
      partsum[bM * 32 + tileN] = vsum;
    }
  }
}

// ---------------------------------------------------------------------------
// Kernel 4: judgement[b] = 1 - max/sum over the 32 tile-partials, then mean.
// ---------------------------------------------------------------------------
__global__ void judge_kernel(const float* __restrict__ partmax,
                             const float* __restrict__ partsum,
                             float* __restrict__ out) {
  __shared__ float red[B_DIM];
  const int b = threadIdx.x;
  float mx = 0.0f, sm = 0.0f;
  for (int t = 0; t < 32; ++t) {
    mx = fmaxf(mx, partmax[b * 32 + t]);
    sm += partsum[b * 32 + t];
  }
  red[b] = 1.0f - mx / sm;
  __syncthreads();
  if (b == 0) {
    float acc = 0.0f;
    for (int t = 0; t < B_DIM; ++t) acc += red[t];
    out[0] = acc / (float)B_DIM;
  }
}

extern "C" void kernel_launch(void* const* d_in, const int* in_sizes, int n_in,
                              void* d_out, int out_size, void* d_ws, size_t ws_size,
                              hipStream_t stream) {
  (void)in_sizes; (void)n_in; (void)out_size;
  const float* attns = (const float*)d_in[0];
  float* ws = (float*)d_ws;

  // i-dimension split for kernel 1: prefer 8-way (2048 blocks -> saturate
  // HBM); fall back to 1 if the workspace is unexpectedly small.
  int isplit = 8;
  size_t need = (size_t)(B_DIM * N_DIM + (size_t)B_DIM * 8 * N_DIM +
                         2 * B_DIM * 32) * sizeof(float);
  if (ws_size < need) isplit = 1;
  const int ilen = N_DIM / isplit;

  float* waves = ws;                                   // B*N floats
  float* part  = ws + B_DIM * N_DIM;                   // B*isplit*N floats
  float* pmax  = part + (size_t)B_DIM * isplit * N_DIM; // B*32 floats
  float* psum  = pmax + B_DIM * 32;                    // B*32 floats

  diag_partial_kernel<<<dim3(N_DIM / 256, isplit, B_DIM), dim3(256), 0, stream>>>(
      attns, part, isplit, ilen);
  waves_finalize_kernel<<<dim3(N_DIM / 256, B_DIM), dim3(256), 0, stream>>>(
      part, waves, isplit);
  dft_wmma_kernel<<<dim3(128), dim3(32), 0, stream>>>(waves, pmax, psum);
  judge_kernel<<<dim3(1), dim3(B_DIM), 0, stream>>>(pmax, psum, (float*)d_out);
}